// NCLModel_15659450761722
// MI455X (gfx1250) — compile-verified
//
#include <hip/hip_runtime.h>
#include <hip/hip_bf16.h>

#define DD      64
#define NU_     4096
#define NI_     4096
#define NE_     262144
#define NPRED_  65536
#define K_CL    1000
#define UNUM    50000
#define INUM    25000
#define IPAD    25008
#define KPAD    1008
#define INV_TEMP 10.0f

typedef float v2f __attribute__((ext_vector_type(2)));
typedef float v8f __attribute__((ext_vector_type(8)));

// ---------------- workspace layout (float offsets) ----------------
#define OFF_EU    ((size_t)0)
#define OFF_EI    (OFF_EU   + (size_t)NU_*DD)
#define OFF_HU    (OFF_EI   + (size_t)NI_*DD)
#define OFF_HI    (OFF_HU   + (size_t)NU_*DD)
#define OFF_NWU   (OFF_HI   + (size_t)NI_*DD)
#define OFF_NWI   (OFF_NWU  + (size_t)NU_*DD)
#define OFF_RU    (OFF_NWI  + (size_t)NI_*DD)
#define OFF_RI    (OFF_RU   + (size_t)NU_*DD)
#define OFF_INVU  (OFF_RI   + (size_t)NI_*DD)
#define OFF_INVI  (OFF_INVU + (size_t)NU_)
#define OFF_NCU   (OFF_INVI + (size_t)NI_)
#define OFF_NPVU  (OFF_NCU  + (size_t)NU_*DD)
#define OFF_NCI   (OFF_NPVU + (size_t)NU_*DD)
#define OFF_NPVI  (OFF_NCI  + (size_t)NI_*DD)
#define OFF_RNU   (OFF_NPVI + (size_t)NI_*DD)
#define OFF_RNI   (OFF_RNU  + (size_t)UNUM)
#define OFF_RCU   (OFF_RNI  + (size_t)IPAD)
#define OFF_RCI   (OFF_RCU  + (size_t)KPAD)
#define OFF_TOTSU (OFF_RCI  + (size_t)KPAD)
#define OFF_TOTSI (OFF_TOTSU + (size_t)NU_)
#define OFF_TOTPU (OFF_TOTSI + (size_t)NI_)
#define OFF_TOTPI (OFF_TOTPU + (size_t)NU_)
#define OFF_SELU  (OFF_TOTPI + (size_t)NI_)
#define OFF_SELI  (OFF_SELU  + (size_t)NU_)
#define OFF_ACC   (OFF_SELI  + (size_t)NI_)

// ---------------- kernels ----------------

__global__ void k_deg(const int* __restrict__ eu, const int* __restrict__ ev,
                      float* __restrict__ degu, float* __restrict__ degi) {
    int t = blockIdx.x * blockDim.x + threadIdx.x;
    if (t >= NE_) return;
    atomicAdd(degu + eu[t], 1.0f);
    atomicAdd(degi + ev[t], 1.0f);
}

__global__ void k_invdeg(float* __restrict__ d, int n) {
    int t = blockIdx.x * blockDim.x + threadIdx.x;
    if (t < n) d[t] = rsqrtf(fmaxf(d[t], 1.0f));
}

__global__ void k_gather(const float* __restrict__ emb, const int* __restrict__ idx,
                         float* __restrict__ e, float* __restrict__ h,
                         float* __restrict__ res, int rows) {
    int t = blockIdx.x * blockDim.x + threadIdx.x;
    if (t >= rows * DD) return;
    int r = t >> 6, d = t & 63;
    float v = emb[(size_t)idx[r] * DD + d];
    e[t] = v; h[t] = v; res[t] = v;
}

// one wave per edge; 2 floats per lane per direction
__global__ void k_edge(const int* __restrict__ eu, const int* __restrict__ ev,
                       const float* __restrict__ hu, const float* __restrict__ hi,
                       const float* __restrict__ invu, const float* __restrict__ invi,
                       float* __restrict__ nwu, float* __restrict__ nwi) {
    int wid = (blockIdx.x * blockDim.x + threadIdx.x) >> 5;
    if (wid >= NE_) return;
    int lane = threadIdx.x & 31;
    int u = eu[wid], v = ev[wid];
    float2 a = *(const float2*)(hu + (size_t)u * DD + lane * 2);
    float su = invu[u];
    atomicAdd(nwi + (size_t)v * DD + lane * 2 + 0, a.x * su);
    atomicAdd(nwi + (size_t)v * DD + lane * 2 + 1, a.y * su);
    float2 b = *(const float2*)(hi + (size_t)v * DD + lane * 2);
    float si = invi[v];
    atomicAdd(nwu + (size_t)u * DD + lane * 2 + 0, b.x * si);
    atomicAdd(nwu + (size_t)u * DD + lane * 2 + 1, b.y * si);
}

__global__ void k_finish(const float* __restrict__ nw, const float* __restrict__ inv,
                         float* __restrict__ h, float* __restrict__ res, int n) {
    int t = blockIdx.x * blockDim.x + threadIdx.x;
    if (t >= n) return;
    float v = nw[t] * inv[t >> 6];
    h[t] = v; res[t] += v;
}

// wave per row: dst = row / max(||row||, 1e-12)
__global__ void k_rownorm(const float* __restrict__ src, float* __restrict__ dst, int rows) {
    int wid = (blockIdx.x * blockDim.x + threadIdx.x) >> 5;
    if (wid >= rows) return;
    int lane = threadIdx.x & 31;
    float2 v = *(const float2*)(src + (size_t)wid * DD + lane * 2);
    float ss = v.x * v.x + v.y * v.y;
    for (int m = 1; m < 32; m <<= 1) ss += __shfl_xor(ss, m, 32);
    float sc = 1.0f / fmaxf(sqrtf(ss), 1e-12f);
    float2 o; o.x = v.x * sc; o.y = v.y * sc;
    *(float2*)(dst + (size_t)wid * DD + lane * 2) = o;
}

// wave per (padded) row: rinv[row] = 1/max(||row||,eps), 0 for pad rows
__global__ void k_rowinv(const float* __restrict__ src, float* __restrict__ rinv,
                         int rows, int pad) {
    int wid = (blockIdx.x * blockDim.x + threadIdx.x) >> 5;
    if (wid >= pad) return;
    int lane = threadIdx.x & 31;
    int r = wid < rows ? wid : rows - 1;
    float2 v = *(const float2*)(src + (size_t)r * DD + lane * 2);
    float ss = v.x * v.x + v.y * v.y;
    for (int m = 1; m < 32; m <<= 1) ss += __shfl_xor(ss, m, 32);
    float sc = 1.0f / fmaxf(sqrtf(ss), 1e-12f);
    if (lane == 0) rinv[wid] = (wid < rows) ? sc : 0.0f;
}

__global__ void k_sel(const int* __restrict__ n2c, const int* __restrict__ idx,
                      int* __restrict__ sel, int n) {
    int t = blockIdx.x * blockDim.x + threadIdx.x;
    if (t < n) sel[t] = n2c[idx[t]];
}

// -------- WMMA GEMM + exp row-sum, double-buffered B fragments --------
// Normalization scale is folded out of the MAC path: lane's 8 C values all
// belong to column lane%16, which is exactly the B row this lane loaded, so
// exp(c * binv[n] * (1/temp)) gives the normalized softmax term.

__device__ __forceinline__ void load_btile(const float* __restrict__ B,
                                           const float* __restrict__ binv,
                                           int t, int l16, int half, int nrowsB,
                                           float2 (&bf)[16], float& scl) {
    int nrow = (t << 4) + l16;
    scl = binv[nrow] * INV_TEMP;
    int nsafe = nrow < nrowsB ? nrow : (nrowsB - 1);   // clamp: keep EXEC full
    const float* brow = B + (size_t)nsafe * DD + 2 * half;
#pragma unroll
    for (int kk = 0; kk < 16; ++kk)
        bf[kk] = *(const float2*)(brow + kk * 4);
}

__device__ __forceinline__ void tile_mma(const float2 (&af)[16], const float2 (&bf)[16],
                                         float scl, float (&racc)[8]) {
    v8f c = {0.f, 0.f, 0.f, 0.f, 0.f, 0.f, 0.f, 0.f};
#pragma unroll
    for (int kk = 0; kk < 16; ++kk) {
        v2f a; a.x = af[kk].x; a.y = af[kk].y;
        v2f b; b.x = bf[kk].x; b.y = bf[kk].y;
        c = __builtin_amdgcn_wmma_f32_16x16x4_f32(false, a, false, b,
                                                  (short)0, c, false, false);
    }
#pragma unroll
    for (int r = 0; r < 8; ++r)
        racc[r] += __expf(c[r] * scl);
}

__global__ void k_gemm_exp_rowsum(const float* __restrict__ A, const float* __restrict__ B,
                                  const float* __restrict__ binv, float* __restrict__ tot,
                                  int nrowsB, int Npad, int tilesPerChunk) {
    const int lane  = threadIdx.x & 31;
    const int half  = lane >> 4;      // K-half select
    const int l16   = lane & 15;
    const int mbase = blockIdx.x * 16;
    const int mrow  = mbase + l16;

    // cache all 16 A k-fragments (K=64, 4 per WMMA)
    float2 afrag[16];
#pragma unroll
    for (int kk = 0; kk < 16; ++kk)
        afrag[kk] = *(const float2*)(A + (size_t)mrow * DD + kk * 4 + 2 * half);

    const int ntiles = Npad >> 4;
    int t0 = blockIdx.y * tilesPerChunk;
    int t1 = t0 + tilesPerChunk; if (t1 > ntiles) t1 = ntiles;
    if (t0 >= t1) return;

    float racc[8];
#pragma unroll
    for (int r = 0; r < 8; ++r) racc[r] = 0.0f;

    float2 bf0[16], bf1[16];
    float s0, s1;
    load_btile(B, binv, t0, l16, half, nrowsB, bf0, s0);

    int t = t0;
    while (t < t1) {
        if (t + 1 < t1) load_btile(B, binv, t + 1, l16, half, nrowsB, bf1, s1);
        tile_mma(afrag, bf0, s0, racc);      // overlaps with bf1 loads in flight
        ++t;
        if (t >= t1) break;
        if (t + 1 < t1) load_btile(B, binv, t + 1, l16, half, nrowsB, bf0, s0);
        tile_mma(afrag, bf1, s1, racc);      // overlaps with bf0 loads in flight
        ++t;
    }

    // C layout: VGPR r, lanes [0,15]=row mbase+r, lanes [16,31]=row mbase+8+r
#pragma unroll
    for (int r = 0; r < 8; ++r) {
        float v = racc[r];
        v += __shfl_xor(v, 1, 16);
        v += __shfl_xor(v, 2, 16);
        v += __shfl_xor(v, 4, 16);
        v += __shfl_xor(v, 8, 16);
        if (l16 == 0) atomicAdd(&tot[mbase + half * 8 + r], v);
    }
}

// wave per row: acc += log(tot[row]-pad) - (A[row]·(B[j]*bs))/temp, j = sel?sel[row]:row
__global__ void k_loss(const float* __restrict__ A, const float* __restrict__ B,
                       const float* __restrict__ binv, const int* __restrict__ sel,
                       const float* __restrict__ tot, float pad,
                       float* __restrict__ acc, int rows) {
    int wid = (blockIdx.x * blockDim.x + threadIdx.x) >> 5;
    if (wid >= rows) return;
    int lane = threadIdx.x & 31;
    int j = sel ? sel[wid] : wid;
    float bs = binv ? binv[j] : 1.0f;
    float2 a = *(const float2*)(A + (size_t)wid * DD + lane * 2);
    float2 b = *(const float2*)(B + (size_t)j * DD + lane * 2);
    float d = a.x * b.x * bs + a.y * b.y * bs;
    for (int m = 1; m < 32; m <<= 1) d += __shfl_xor(d, m, 32);
    if (lane == 0) atomicAdd(acc, logf(tot[wid] - pad) - d * INV_TEMP);
}

__global__ void k_scores(const float* __restrict__ ru, const float* __restrict__ ri,
                         const int* __restrict__ pu, const int* __restrict__ pv,
                         const int* __restrict__ nu, const int* __restrict__ nv,
                         float* __restrict__ out) {
    int t = blockIdx.x * blockDim.x + threadIdx.x;
    if (t >= 2 * NPRED_) return;
    int u, v;
    if (t < NPRED_) { u = pu[t]; v = pv[t]; }
    else            { u = nu[t - NPRED_]; v = nv[t - NPRED_]; }
    const float4* a = (const float4*)(ru + (size_t)u * DD);
    const float4* b = (const float4*)(ri + (size_t)v * DD);
    float s = 0.0f;
#pragma unroll
    for (int q = 0; q < 16; ++q) {
        float4 x = a[q], y = b[q];
        s += x.x * y.x + x.y * y.y + x.z * y.z + x.w * y.w;
    }
    out[t] = s * (1.0f / 16.0f);   // res_u/4 · res_i/4
}

__global__ void k_final(const float* __restrict__ acc, float* __restrict__ out) {
    if (threadIdx.x == 0 && blockIdx.x == 0) {
        out[2 * NPRED_]     = 1e-6f * (acc[0] + 1.0f * acc[1]);  // SSL_REG*(u + ALPHA*i)
        out[2 * NPRED_ + 1] = 8e-8f * (acc[2] + acc[3]);         // PROTO_REG*(u + i)
    }
}

// ---------------- host ----------------
static inline int cdiv(int a, int b) { return (a + b - 1) / b; }

extern "C" void kernel_launch(void* const* d_in, const int* in_sizes, int n_in,
                              void* d_out, int out_size, void* d_ws, size_t ws_size,
                              hipStream_t stream) {
    (void)in_sizes; (void)n_in; (void)out_size; (void)ws_size;
    const float* emb_user = (const float*)d_in[0];
    const float* emb_item = (const float*)d_in[1];
    const float* ucent    = (const float*)d_in[2];
    const float* icent    = (const float*)d_in[3];
    const int* edge_u = (const int*)d_in[4];
    const int* edge_v = (const int*)d_in[5];
    const int* pos_u  = (const int*)d_in[6];
    const int* pos_v  = (const int*)d_in[7];
    const int* neg_u  = (const int*)d_in[8];
    const int* neg_v  = (const int*)d_in[9];
    const int* idx_user = (const int*)d_in[10];
    const int* idx_item = (const int*)d_in[11];
    const int* u2c = (const int*)d_in[12];
    const int* i2c = (const int*)d_in[13];
    float* out = (float*)d_out;

    float* ws   = (float*)d_ws;
    float* e_u  = ws + OFF_EU;   float* e_i  = ws + OFF_EI;
    float* h_u  = ws + OFF_HU;   float* h_i  = ws + OFF_HI;
    float* nw_u = ws + OFF_NWU;  float* nw_i = ws + OFF_NWI;
    float* r_u  = ws + OFF_RU;   float* r_i  = ws + OFF_RI;
    float* invu = ws + OFF_INVU; float* invi = ws + OFF_INVI;
    float* nc_u = ws + OFF_NCU;  float* npv_u= ws + OFF_NPVU;
    float* nc_i = ws + OFF_NCI;  float* npv_i= ws + OFF_NPVI;
    float* rnu  = ws + OFF_RNU;  float* rni  = ws + OFF_RNI;
    float* rcu  = ws + OFF_RCU;  float* rci  = ws + OFF_RCI;
    float* tsu  = ws + OFF_TOTSU;float* tsi  = ws + OFF_TOTSI;
    float* tpu  = ws + OFF_TOTPU;float* tpi  = ws + OFF_TOTPI;
    int*   selu = (int*)(ws + OFF_SELU);
    int*   seli = (int*)(ws + OFF_SELI);
    float* acc  = ws + OFF_ACC;

    hipMemsetAsync(invu, 0, (size_t)(NU_ + NI_) * sizeof(float), stream);
    hipMemsetAsync(tsu,  0, (size_t)4 * NU_ * sizeof(float), stream);
    hipMemsetAsync(acc,  0, 4 * sizeof(float), stream);

    // degrees -> 1/sqrt(max(deg,1))
    k_deg<<<cdiv(NE_, 256), 256, 0, stream>>>(edge_u, edge_v, invu, invi);
    k_invdeg<<<cdiv(NU_ + NI_, 256), 256, 0, stream>>>(invu, NU_ + NI_);

    // gather initial sub-embeddings
    k_gather<<<cdiv(NU_ * DD, 256), 256, 0, stream>>>(emb_user, idx_user, e_u, h_u, r_u, NU_);
    k_gather<<<cdiv(NI_ * DD, 256), 256, 0, stream>>>(emb_item, idx_item, e_i, h_i, r_i, NI_);

    // 3 GraphConv layers
    for (int l = 0; l < 3; ++l) {
        hipMemsetAsync(nw_u, 0, (size_t)(NU_ + NI_) * DD * sizeof(float), stream);
        k_edge<<<(NE_ * 32) / 256, 256, 0, stream>>>(edge_u, edge_v, h_u, h_i,
                                                     invu, invi, nw_u, nw_i);
        k_finish<<<cdiv(NU_ * DD, 256), 256, 0, stream>>>(nw_u, invu, h_u, r_u, NU_ * DD);
        k_finish<<<cdiv(NI_ * DD, 256), 256, 0, stream>>>(nw_i, invi, h_i, r_i, NI_ * DD);
    }

    // row-normalized batch matrices
    k_rownorm<<<cdiv(NU_, 8), 256, 0, stream>>>(h_u, nc_u,  NU_);
    k_rownorm<<<cdiv(NU_, 8), 256, 0, stream>>>(e_u, npv_u, NU_);
    k_rownorm<<<cdiv(NI_, 8), 256, 0, stream>>>(h_i, nc_i,  NI_);
    k_rownorm<<<cdiv(NI_, 8), 256, 0, stream>>>(e_i, npv_i, NI_);

    // table inverse norms (pad rows -> 0)
    k_rowinv<<<cdiv(UNUM, 8), 256, 0, stream>>>(emb_user, rnu, UNUM, UNUM);
    k_rowinv<<<cdiv(IPAD, 8), 256, 0, stream>>>(emb_item, rni, INUM, IPAD);
    k_rowinv<<<cdiv(KPAD, 8), 256, 0, stream>>>(ucent, rcu, K_CL, KPAD);
    k_rowinv<<<cdiv(KPAD, 8), 256, 0, stream>>>(icent, rci, K_CL, KPAD);

    // proto cluster selectors
    k_sel<<<cdiv(NU_, 256), 256, 0, stream>>>(u2c, idx_user, selu, NU_);
    k_sel<<<cdiv(NI_, 256), 256, 0, stream>>>(i2c, idx_item, seli, NI_);

    // WMMA softmax denominators
    const int TPC = 32;
    k_gemm_exp_rowsum<<<dim3(NU_ / 16, cdiv(UNUM / 16, TPC)), 32, 0, stream>>>(
        nc_u, emb_user, rnu, tsu, UNUM, UNUM, TPC);
    k_gemm_exp_rowsum<<<dim3(NI_ / 16, cdiv(IPAD / 16, TPC)), 32, 0, stream>>>(
        nc_i, emb_item, rni, tsi, INUM, IPAD, TPC);
    k_gemm_exp_rowsum<<<dim3(NU_ / 16, cdiv(KPAD / 16, TPC)), 32, 0, stream>>>(
        npv_u, ucent, rcu, tpu, K_CL, KPAD, TPC);
    k_gemm_exp_rowsum<<<dim3(NI_ / 16, cdiv(KPAD / 16, TPC)), 32, 0, stream>>>(
        npv_i, icent, rci, tpi, K_CL, KPAD, TPC);

    // loss accumulation (pad correction: exp(0)=1 per padded column)
    k_loss<<<cdiv(NU_, 8), 256, 0, stream>>>(nc_u, npv_u, nullptr, nullptr, tsu, 0.0f, acc + 0, NU_);
    k_loss<<<cdiv(NI_, 8), 256, 0, stream>>>(nc_i, npv_i, nullptr, nullptr, tsi, 8.0f, acc + 1, NI_);
    k_loss<<<cdiv(NU_, 8), 256, 0, stream>>>(npv_u, ucent, rcu, selu, tpu, 8.0f, acc + 2, NU_);
    k_loss<<<cdiv(NI_, 8), 256, 0, stream>>>(npv_i, icent, rci, seli, tpi, 8.0f, acc + 3, NI_);

    // predictor scores + scalar outputs
    k_scores<<<cdiv(2 * NPRED_, 256), 256, 0, stream>>>(r_u, r_i, pos_u, pos_v, neg_u, neg_v, out);
    k_final<<<1, 1, 0, stream>>>(acc, out);
}